// GraphFilteringLayer_62895501082684
// MI455X (gfx1250) — compile-verified
//
#include <hip/hip_runtime.h>

typedef __attribute__((ext_vector_type(2))) float v2f;
typedef __attribute__((ext_vector_type(8))) float v8f;

#define N_NODES 100000
#define N_EDGES 3200000
#define F_IN    64
#define F_OUT   64
#define ORDER   4
#define P_MAX   193   // (F_OUT-1)*(ORDER-1) + ORDER
#define CHUNK   8     // edges per thread in SpMV (N_EDGES % CHUNK == 0)

// ---------------------------------------------------------------------------
// Zero the [N_NODES, F_OUT] output accumulator (d_out is poisoned by harness).
// ---------------------------------------------------------------------------
__global__ void zero_out_kernel(float* __restrict__ out) {
  int i = blockIdx.x * blockDim.x + threadIdx.x;
  const int n4 = (N_NODES * F_OUT) / 4;   // 1.6M float4
  if (i < n4) {
    float4* o4 = reinterpret_cast<float4*>(out);
    o4[i] = make_float4(0.f, 0.f, 0.f, 0.f);
  }
}

// ---------------------------------------------------------------------------
// v0[n] = sum_f x[n,f]  via V_WMMA_F32_16X16X4_F32 with an all-ones B matrix.
// One wave handles a 16-node tile; 16 chained K=4 WMMAs accumulate the 64-wide
// feature sum into C. Column N=0 of C is extracted per the CDNA5 C/D layout:
//   VGPR r: lanes 0-15 -> M=r (N=lane), lanes 16-31 -> M=8+r (N=lane-16).
// A layout (32-bit A 16x4): lanes 0-15 hold K=0,1 in VGPR0,1; lanes 16-31 K=2,3.
// ---------------------------------------------------------------------------
__global__ void v0_wmma_kernel(const float* __restrict__ x, float* __restrict__ v0) {
  int gtid = blockIdx.x * blockDim.x + threadIdx.x;
  int wave = gtid >> 5;                       // wave32
  int lane = threadIdx.x & 31;
  const int NWAVES = N_NODES / 16;            // 6250 tiles, exact
  if (wave >= NWAVES) return;                 // whole-wave guard: EXEC stays all-1s

  int base  = wave * 16;
  int m     = lane & 15;                      // row within tile
  int khalf = lane >> 4;                      // 0: K=0,1   1: K=2,3
  const float* rowp = x + (size_t)(base + m) * F_IN + khalf * 2;

  v2f b; b[0] = 1.0f; b[1] = 1.0f;            // ones matrix (layout-invariant)
  v8f c = {};
#pragma unroll
  for (int j = 0; j < F_IN / 4; ++j) {
    v2f a;
    a[0] = rowp[j * 4 + 0];
    a[1] = rowp[j * 4 + 1];
    // 8 args: (neg_a, A, neg_b, B, c_mod, C, reuse_a, reuse_b)
    c = __builtin_amdgcn_wmma_f32_16x16x4_f32(false, a, false, b,
                                              (short)0, c, false, false);
  }
  if (lane == 0) {
#pragma unroll
    for (int r = 0; r < 8; ++r) v0[base + r] = c[r];
  } else if (lane == 16) {
#pragma unroll
    for (int r = 0; r < 8; ++r) v0[base + 8 + r] = c[r];
  }
}

// ---------------------------------------------------------------------------
// Per-iteration prep: zero the SpMV destination buffer AND fold the previous
// power Z[p-1] (= src) into the output. Power q = (p-1)-1 = f*3+i has at most
// two valid (f,i) pairs, so Z never needs to be materialized beyond ping-pong.
// ---------------------------------------------------------------------------
__global__ void prep_kernel(const float* __restrict__ src, float* __restrict__ dst,
                            float* __restrict__ out, const float* __restrict__ coeff,
                            int p) {
  int n = blockIdx.x * blockDim.x + threadIdx.x;
  if (n >= N_NODES) return;
  dst[n] = 0.0f;                              // fresh accumulator for next SpMV
  int q = p - 2;                              // tap index f*3+i of Z[p-1]
  if (q < 0) return;                          // p==1: nothing to combine yet
  float v = src[n];
  int fmin = (q <= 3) ? 0 : (q - 1) / 3;      // ceil((q-3)/3) clamped to 0
  int fmax = q / 3; if (fmax > F_OUT - 1) fmax = F_OUT - 1;
  for (int f = fmin; f <= fmax; ++f) {        // 1 or 2 iterations
    int i = q - 3 * f;                        // 0..3 by construction
    out[(size_t)n * F_OUT + f] += coeff[f * ORDER + i] * v;
  }
}

// ---------------------------------------------------------------------------
// One SpMV step: dst[row[e]] += vals[e] * src[col[e]].
// Each thread owns 8 contiguous edges (float4/int4 loads -> coalesced across
// the wave). Rows are sorted, so run-length accumulate and flush one
// global_atomic_add_f32 per row change (~1.3 atomics per 8 edges on average).
// Gather src[col] and all edge data stay resident in the 192 MB L2.
// ---------------------------------------------------------------------------
__global__ void spmv_kernel(const float* __restrict__ vals, const int* __restrict__ col,
                            const int* __restrict__ row, const float* __restrict__ src,
                            float* __restrict__ dst) {
  int t = blockIdx.x * blockDim.x + threadIdx.x;
  int e0 = t * CHUNK;
  if (e0 >= N_EDGES) return;

  const float4* v4 = reinterpret_cast<const float4*>(vals + e0);
  const int4*   c4 = reinterpret_cast<const int4*>(col + e0);
  const int4*   r4 = reinterpret_cast<const int4*>(row + e0);
  float4 va0 = v4[0], va1 = v4[1];
  int4   cc0 = c4[0], cc1 = c4[1];
  int4   rr0 = r4[0], rr1 = r4[1];

  float va[CHUNK] = {va0.x, va0.y, va0.z, va0.w, va1.x, va1.y, va1.z, va1.w};
  int   cl[CHUNK] = {cc0.x, cc0.y, cc0.z, cc0.w, cc1.x, cc1.y, cc1.z, cc1.w};
  int   rw[CHUNK] = {rr0.x, rr0.y, rr0.z, rr0.w, rr1.x, rr1.y, rr1.z, rr1.w};

  float g[CHUNK];
#pragma unroll
  for (int k = 0; k < CHUNK; ++k) g[k] = src[cl[k]];  // issue gathers early

  float acc = 0.0f;
  int   cur = rw[0];
#pragma unroll
  for (int k = 0; k < CHUNK; ++k) {
    if (rw[k] != cur) {
      atomicAdd(&dst[cur], acc);
      cur = rw[k];
      acc = 0.0f;
    }
    acc += va[k] * g[k];
  }
  atomicAdd(&dst[cur], acc);
}

// ---------------------------------------------------------------------------
extern "C" void kernel_launch(void* const* d_in, const int* in_sizes, int n_in,
                              void* d_out, int out_size, void* d_ws, size_t ws_size,
                              hipStream_t stream) {
  const float* x     = (const float*)d_in[0];   // [N_NODES, F_IN]
  const float* coeff = (const float*)d_in[1];   // [F_OUT, ORDER]
  const float* vals  = (const float*)d_in[2];   // [N_EDGES]
  const int*   row   = (const int*)d_in[3];     // [N_EDGES], sorted
  const int*   col   = (const int*)d_in[4];     // [N_EDGES]
  // d_in[5] = ind (unused by the reference math)
  float* out = (float*)d_out;                   // [N_NODES, F_OUT]

  // Ping-pong vectors in workspace (2 x ~400KB, 256B-aligned stride).
  float* bufA = (float*)d_ws;
  float* bufB = bufA + 100352;
  float* bufs[2] = {bufA, bufB};

  // 1) zero output accumulator
  zero_out_kernel<<<(N_NODES * F_OUT / 4 + 255) / 256, 256, 0, stream>>>(out);

  // 2) v0 = x @ ones  (WMMA f32 16x16x4), written to bufA = Z[0]
  v0_wmma_kernel<<<((N_NODES / 16) * 32 + 255) / 256, 256, 0, stream>>>(x, bufA);

  // 3) chain of 193 SpMVs; prep(p) zeroes dst and folds Z[p-1] into out.
  //    Final prep (p = P_MAX+1) folds Z[193]; its dst-zero is harmless.
  const int prep_grid = (N_NODES + 255) / 256;
  const int spmv_grid = (N_EDGES / CHUNK + 255) / 256;
  for (int p = 1; p <= P_MAX + 1; ++p) {
    float* src = bufs[(p - 1) & 1];
    float* dst = bufs[p & 1];
    prep_kernel<<<prep_grid, 256, 0, stream>>>(src, dst, out, coeff, p);
    if (p <= P_MAX)
      spmv_kernel<<<spmv_grid, 256, 0, stream>>>(vals, col, row, src, dst);
  }
}